// GlowPrediction_9148280340593
// MI455X (gfx1250) — compile-verified
//
#include <hip/hip_runtime.h>

// ---------------------------------------------------------------------------
// GlowPrediction on MI455X (gfx1250, wave32).
//   rhsT = O^T @ Y^T                      (f32 WMMA GEMM, transposed reads)
//   (U U^T) x0 = rhsT  (two triangular solves, one wave per column)
//   CA = C @ A                            (f32 WMMA GEMM)
//   y0 = C @ x0                           (f32 WMMA GEMM)
//   for t = 1..T-1:  x_t = A @ x_{t-1};  y_t = CA @ x_{t-1}   (fused launch)
// All matrix math uses V_WMMA_F32_16X16X4_F32 (fp32 needed: the 255-step
// recursion amplifies input-rounding error by ~1/(1-0.95) ~ 20x, ruling out
// bf16/f16 inputs against an fp32 reference).
// Each wave computes a 32x64 tile: 2 A-frags x 4 B-frags -> 8 WMMAs per
// k-step against 2 b64 + 8 b32 loads (~1.25 vmem/wmma vs 2.25 for 16x64).
// ---------------------------------------------------------------------------

typedef float v2f __attribute__((ext_vector_type(2)));
typedef float v8f __attribute__((ext_vector_type(8)));

static __device__ __forceinline__ v8f wmma_f32(v2f a, v2f b, v8f c) {
  // D = A(16x4, f32) * B(4x16, f32) + C(16x16, f32)
  return __builtin_amdgcn_wmma_f32_16x16x4_f32(
      /*neg_a=*/false, a, /*neg_b=*/false, b,
      /*c_mod=*/(short)0, c, /*reuse_a=*/false, /*reuse_b=*/false);
}

// One wave computes a 32(M) x 64(N) tile of C = op(A) @ op(B), K in steps of 4.
// A-frag layout (ISA 7.12.2, 32-bit A 16x4): lane<16 -> K=0,1 ; lane>=16 -> K=2,3.
// B-frag mirrors it: lane<16 -> rows K=0,1 at col N=lane ; lane>=16 -> K=2,3.
// C/D layout: VGPR r, lanes 0-15 -> M=r, lanes 16-31 -> M=8+r, N=lane&15.
template <bool TA, bool TB>
static __device__ __forceinline__ void wave_gemm_32x64(
    const float* __restrict__ A, int lda,
    const float* __restrict__ B, int ldb,
    float* __restrict__ C, int ldc,
    int K, int m0, int n0) {
  const int lane = threadIdx.x & 31;
  const int ll = lane & 15;   // M (for A) / N (for B,C) within tile
  const int lh = lane >> 4;   // selects K pair {0,1} vs {2,3}
  const int row0 = m0 + ll;        // top 16-row half
  const int row1 = m0 + 16 + ll;   // bottom 16-row half

  v8f z = {0.f, 0.f, 0.f, 0.f, 0.f, 0.f, 0.f, 0.f};
  v8f acc00 = z, acc01 = z, acc02 = z, acc03 = z;  // rows [m0, m0+16)
  v8f acc10 = z, acc11 = z, acc12 = z, acc13 = z;  // rows [m0+16, m0+32)

  for (int k = 0; k < K; k += 4) {
    const int kk = k + 2 * lh;
    v2f a0, a1;
    if (TA) {  // op(A)[m][k] = A[k][m]
      a0.x = A[(size_t)kk * lda + row0];
      a0.y = A[(size_t)(kk + 1) * lda + row0];
      a1.x = A[(size_t)kk * lda + row1];
      a1.y = A[(size_t)(kk + 1) * lda + row1];
    } else {   // row-major A: contiguous pair -> b64 load
      const float* ap0 = A + (size_t)row0 * lda + kk;
      const float* ap1 = A + (size_t)row1 * lda + kk;
      a0.x = ap0[0]; a0.y = ap0[1];
      a1.x = ap1[0]; a1.y = ap1[1];
    }
    v2f b0, b1, b2, b3;
#define LOADB(bj, j)                                                  \
    {                                                                 \
      const int col = n0 + 16 * (j) + ll;                             \
      if (TB) { /* op(B)[k][n] = B[n][k]: contiguous pair */          \
        const float* bp = B + (size_t)col * ldb + kk;                 \
        bj.x = bp[0];                                                 \
        bj.y = bp[1];                                                 \
      } else {                                                        \
        bj.x = B[(size_t)kk * ldb + col];                             \
        bj.y = B[(size_t)(kk + 1) * ldb + col];                       \
      }                                                               \
    }
    LOADB(b0, 0) LOADB(b1, 1) LOADB(b2, 2) LOADB(b3, 3)
#undef LOADB
    acc00 = wmma_f32(a0, b0, acc00);
    acc01 = wmma_f32(a0, b1, acc01);
    acc02 = wmma_f32(a0, b2, acc02);
    acc03 = wmma_f32(a0, b3, acc03);
    acc10 = wmma_f32(a1, b0, acc10);
    acc11 = wmma_f32(a1, b1, acc11);
    acc12 = wmma_f32(a1, b2, acc12);
    acc13 = wmma_f32(a1, b3, acc13);
  }

#pragma unroll
  for (int r = 0; r < 8; ++r) {
    float* cp0 = C + (size_t)(m0 + r + 8 * lh) * ldc + n0 + ll;
    cp0[0]  = acc00[r];
    cp0[16] = acc01[r];
    cp0[32] = acc02[r];
    cp0[48] = acc03[r];
    float* cp1 = C + (size_t)(m0 + 16 + r + 8 * lh) * ldc + n0 + ll;
    cp1[0]  = acc10[r];
    cp1[16] = acc11[r];
    cp1[32] = acc12[r];
    cp1[48] = acc13[r];
  }
}

// Block = 256 threads = 8 waves arranged 4(M) x 2(N) -> 128 x 128 block tile.
template <bool TA, bool TB>
__global__ void __launch_bounds__(256)
wmma_gemm_kernel(const float* __restrict__ A, int lda,
                 const float* __restrict__ B, int ldb,
                 float* __restrict__ C, int ldc, int K) {
  const int wave = threadIdx.x >> 5;
  const int m0 = blockIdx.x * 128 + (wave & 3) * 32;
  const int n0 = blockIdx.y * 128 + (wave >> 2) * 64;
  wave_gemm_32x64<TA, TB>(A, lda, B, ldb, C, ldc, K, m0, n0);
}

// One launch per time step: grid.z = 0 -> x_t = A @ x_{t-1}
//                           grid.z = 1 -> y_t = (C A) @ x_{t-1}
// Both depend only on x_{t-1}, so a single launch has no internal dependency.
__global__ void __launch_bounds__(256)
step_kernel(const float* __restrict__ Amat, const float* __restrict__ CAmat,
            const float* __restrict__ xin, float* __restrict__ xout,
            float* __restrict__ yout) {
  const float* W = (blockIdx.z == 0) ? Amat : CAmat;
  float* O = (blockIdx.z == 0) ? xout : yout;
  const int wave = threadIdx.x >> 5;
  const int m0 = blockIdx.x * 128 + (wave & 3) * 32;
  const int n0 = blockIdx.y * 128 + (wave >> 2) * 64;
  wave_gemm_32x64<false, false>(W, 256, xin, 1024, O, 1024, 256, m0, n0);
}

// Solve (U U^T) X = RHS for 1024 columns. One wave32 per column.
// The 256-length solution vector lives register-distributed: lane l holds
// indices {l, l+32, ..., l+224}. Full-row dot products are exact because
// U is strictly lower-triangular above the diagonal and unset entries are 0.
__global__ void __launch_bounds__(256)
trsolve_kernel(const float* __restrict__ U, const float* __restrict__ rhsT,
               float* __restrict__ X, int mcols) {
  const int wave = threadIdx.x >> 5;
  const int lane = threadIdx.x & 31;
  const int col = blockIdx.x * 8 + wave;

  float z[8], x[8];
#pragma unroll
  for (int r = 0; r < 8; ++r) { z[r] = 0.f; x[r] = 0.f; }

  // Forward: U z = rhsT[:, col]
#pragma unroll
  for (int r = 0; r < 8; ++r) {
    for (int ii = 0; ii < 32; ++ii) {
      const int i = 32 * r + ii;
      float s = 0.f;
#pragma unroll
      for (int q = 0; q < 8; ++q)
        s += U[(size_t)i * 256 + lane + 32 * q] * z[q];
#pragma unroll
      for (int m = 16; m >= 1; m >>= 1) s += __shfl_xor(s, m);
      const float zi = (rhsT[(size_t)i * mcols + col] - s) / U[(size_t)i * 256 + i];
      if (lane == ii) z[r] = zi;
    }
  }

  // Backward: U^T x = z   (column reads of U; upper part of U^T)
#pragma unroll
  for (int r = 7; r >= 0; --r) {
    for (int ii = 31; ii >= 0; --ii) {
      const int i = 32 * r + ii;
      float s = 0.f;
#pragma unroll
      for (int q = 0; q < 8; ++q)
        s += U[(size_t)(lane + 32 * q) * 256 + i] * x[q];
#pragma unroll
      for (int m = 16; m >= 1; m >>= 1) s += __shfl_xor(s, m);
      const float zv = __shfl(z[r], ii);
      const float xi = (zv - s) / U[(size_t)i * 256 + i];
      if (lane == ii) x[r] = xi;
    }
  }

#pragma unroll
  for (int r = 0; r < 8; ++r)
    X[(size_t)(lane + 32 * r) * mcols + col] = x[r];
}

extern "C" void kernel_launch(void* const* d_in, const int* in_sizes, int n_in,
                              void* d_out, int out_size, void* d_ws, size_t ws_size,
                              hipStream_t stream) {
  (void)in_sizes; (void)n_in; (void)ws_size;
  const float* Y  = (const float*)d_in[0];  // [1024, 256]
  const float* O  = (const float*)d_in[1];  // [256, 256]
  const float* U  = (const float*)d_in[2];  // [256, 256] lower-tri
  const float* A  = (const float*)d_in[3];  // [256, 256]
  const float* Cm = (const float*)d_in[4];  // [256, 256]

  const int NN = 256, PP = 256, MM = 1024;
  const int T = out_size / (PP * MM);       // 256; avoids reading device scalar

  float* out  = (float*)d_out;
  float* ws   = (float*)d_ws;
  float* rhsT = ws;                 // 256*1024 floats
  float* xb0  = ws + 262144;        // 256*1024
  float* xb1  = ws + 524288;        // 256*1024
  float* CA   = ws + 786432;        // 256*256

  const dim3 blk(256);

  // rhsT[n][m] = (Y @ O)^T = O^T @ Y^T   (transposed reads of O and Y)
  wmma_gemm_kernel<true, true><<<dim3(2, 8), blk, 0, stream>>>(
      O, NN, Y, PP, rhsT, MM, PP);

  // x0 = (U U^T)^{-1} rhsT
  trsolve_kernel<<<dim3(MM / 8), blk, 0, stream>>>(U, rhsT, xb0, MM);

  // CA = C @ A
  wmma_gemm_kernel<false, false><<<dim3(2, 2), blk, 0, stream>>>(
      Cm, NN, A, NN, CA, NN, NN);

  // y0 = C @ x0
  wmma_gemm_kernel<false, false><<<dim3(2, 8), blk, 0, stream>>>(
      Cm, NN, xb0, MM, out, MM, NN);

  // Sequential rollout: stream order provides the t -> t+1 dependency.
  const float* cur = xb0;
  float* nxt = xb1;
  for (int t = 1; t < T; ++t) {
    step_kernel<<<dim3(2, 8, 2), blk, 0, stream>>>(
        A, CA, cur, nxt, out + (size_t)t * PP * MM);
    const float* tmp = nxt;
    nxt = (float*)cur;
    cur = tmp;
  }
}